// NF_24859270709926
// MI455X (gfx1250) — compile-verified
//
#include <hip/hip_runtime.h>
#include <hip/hip_bf16.h>
#include <math.h>

#define NN   50000
#define EE   800000
#define HDIM 64
#define ASTRIDE 68   // padded LDS stride: 68 % 64 == 4 -> conflict-free 16-row fragments

#define ACT_NONE 0
#define ACT_RELU 1
#define ACT_SIG  2

typedef __attribute__((ext_vector_type(2))) float v2f;
typedef __attribute__((ext_vector_type(8))) float v8f;

// ---------------- setup kernels ----------------

// Split x (N x 128) into x0/x1 (N x 64 each).
__global__ void k_split_x(const float* __restrict__ x, float* __restrict__ x0,
                          float* __restrict__ x1, int n) {
  int tid = blockIdx.x * blockDim.x + threadIdx.x;
  if (tid >= n * HDIM) return;
  int node = tid / HDIM, c = tid % HDIM;
  x0[tid] = x[node * 2 * HDIM + c];
  x1[tid] = x[node * 2 * HDIM + HDIM + c];
}

// deg[n] = 1 (self loop); zero the ldj accumulator once.
__global__ void k_deg_init(float* __restrict__ deg, float* __restrict__ ldj, int n) {
  int tid = blockIdx.x * blockDim.x + threadIdx.x;
  if (tid == 0) *ldj = 0.0f;
  if (tid < n) deg[tid] = 1.0f;
}

__global__ void k_deg_edges(const int* __restrict__ dst, float* __restrict__ deg, int e) {
  int tid = blockIdx.x * blockDim.x + threadIdx.x;
  if (tid < e) atomicAdd(&deg[dst[tid]], 1.0f);
}

// In-place: deg -> dinv = rsqrt(max(deg,1))
__global__ void k_dinv(float* __restrict__ deg, int n) {
  int tid = blockIdx.x * blockDim.x + threadIdx.x;
  if (tid < n) deg[tid] = rsqrtf(fmaxf(deg[tid], 1.0f));
}

// ---------------- propagation: P = A_norm * h ----------------

// Self-loop term doubles as the zero-init of P: P[i] = h[i] * dinv[i]^2
__global__ void k_prop_self(const float* __restrict__ h, const float* __restrict__ dinv,
                            float* __restrict__ P, int n) {
  int tid = blockIdx.x * blockDim.x + threadIdx.x;
  if (tid >= n * HDIM) return;
  float di = dinv[tid >> 6];
  P[tid] = h[tid] * di * di;
}

// 16 threads per edge, float4 per thread: gather h[src], scale, scatter-add to P[dst].
// Whole working set (~100 MB) is L2-resident (192 MB), so fp32 L2 atomics are cheap.
__global__ void k_prop_edges(const float* __restrict__ h, const int* __restrict__ src,
                             const int* __restrict__ dst, const float* __restrict__ dinv,
                             float* __restrict__ P, int e) {
  int tid = blockIdx.x * blockDim.x + threadIdx.x;
  int eid = tid >> 4;
  if (eid >= e) return;
  int f = (tid & 15) << 2;
  int s = src[eid];
  int d = dst[eid];
  float w = dinv[s] * dinv[d];
  const float4 v = *reinterpret_cast<const float4*>(h + (size_t)s * HDIM + f);
  float* pd = P + (size_t)d * HDIM + f;
  atomicAdd(pd + 0, v.x * w);
  atomicAdd(pd + 1, v.y * w);
  atomicAdd(pd + 2, v.z * w);
  atomicAdd(pd + 3, v.w * w);
}

// ---------------- WMMA GEMM: out = act(in @ W + bias) ----------------
// Block = 256 threads = 8 waves; block covers 128 rows x 64 cols.
// Each wave: one 16-row slab, all 4 col tiles -> 4 independent accumulation
// chains (covers WMMA->WMMA RAW latency), 4x A-fragment reuse, 64 WMMAs/wave.
// A tile AND W^T staged in LDS with padded stride so every operand pair
// (k, k+1) is a contiguous, conflict-free ds_load_b64 -> no repack moves.
__global__ void __launch_bounds__(256)
k_gemm64(const float* __restrict__ in, const float* __restrict__ W,
         const float* __restrict__ bias, float* __restrict__ out, int n, int act) {
  __shared__ float sWt[64 * ASTRIDE];    // W transposed: sWt[col*ASTRIDE + k]
  __shared__ float sB[64];
  __shared__ float sA[128 * ASTRIDE];

  // stage W transposed (so B fragment pairs are contiguous in LDS)
  {
    const float4* W4 = reinterpret_cast<const float4*>(W);
#pragma unroll
    for (int i = 0; i < 4; i++) {
      int idx = threadIdx.x + i * 256;   // 0..1023
      int k   = idx >> 4;                // W row (K index)
      int c4  = (idx & 15) << 2;         // W col
      float4 v = W4[idx];
      sWt[(c4 + 0) * ASTRIDE + k] = v.x;
      sWt[(c4 + 1) * ASTRIDE + k] = v.y;
      sWt[(c4 + 2) * ASTRIDE + k] = v.z;
      sWt[(c4 + 3) * ASTRIDE + k] = v.w;
    }
  }
  if (threadIdx.x < 64) sB[threadIdx.x] = bias[threadIdx.x];

  // stage A tile: 128 rows x 64 cols -> LDS (padded stride), zero-fill OOB rows
  const int rowsbase = blockIdx.x * 128;
  {
    const float4* in4 = reinterpret_cast<const float4*>(in);
#pragma unroll
    for (int i = 0; i < 8; i++) {
      int f4  = threadIdx.x + i * 256;  // 0..2047 (128 rows * 16 float4)
      int rw  = f4 >> 4;
      int c4  = f4 & 15;
      int row = rowsbase + rw;
      float4 v = make_float4(0.0f, 0.0f, 0.0f, 0.0f);
      if (row < n) v = in4[(size_t)row * 16 + c4];
      *reinterpret_cast<float4*>(&sA[rw * ASTRIDE + c4 * 4]) = v;
    }
  }
  __syncthreads();

  const int wave = threadIdx.x >> 5;
  const int lane = threadIdx.x & 31;
  const int r    = lane & 15;
  const int koff = (lane >> 4) << 1;  // A/B 16x4 layout: lanes 0-15 K={0,1}, 16-31 K={2,3}
  const float* a  = &sA[(wave * 16 + r) * ASTRIDE];
  const float* w0 = &sWt[(r)      * ASTRIDE];
  const float* w1 = &sWt[(16 + r) * ASTRIDE];
  const float* w2 = &sWt[(32 + r) * ASTRIDE];
  const float* w3 = &sWt[(48 + r) * ASTRIDE];

  v8f acc0 = {}, acc1 = {}, acc2 = {}, acc3 = {};
#pragma unroll
  for (int k = 0; k < 64; k += 4) {
    int kk = k + koff;
    v2f av, b0, b1, b2, b3;
    av.x = a[kk];  av.y = a[kk + 1];
    b0.x = w0[kk]; b0.y = w0[kk + 1];
    b1.x = w1[kk]; b1.y = w1[kk + 1];
    b2.x = w2[kk]; b2.y = w2[kk + 1];
    b3.x = w3[kk]; b3.y = w3[kk + 1];
    acc0 = __builtin_amdgcn_wmma_f32_16x16x4_f32(false, av, false, b0, (short)0, acc0, false, false);
    acc1 = __builtin_amdgcn_wmma_f32_16x16x4_f32(false, av, false, b1, (short)0, acc1, false, false);
    acc2 = __builtin_amdgcn_wmma_f32_16x16x4_f32(false, av, false, b2, (short)0, acc2, false, false);
    acc3 = __builtin_amdgcn_wmma_f32_16x16x4_f32(false, av, false, b3, (short)0, acc3, false, false);
  }

  // C/D layout: VGPR i -> M = i (lanes 0-15) or i+8 (lanes 16-31), N = lane%16
  const int mbase = rowsbase + wave * 16 + ((lane >> 4) << 3);
  const float bc0 = sB[r], bc1 = sB[16 + r], bc2 = sB[32 + r], bc3 = sB[48 + r];
#pragma unroll
  for (int i = 0; i < 8; i++) {
    int m = mbase + i;
    if (m < n) {
      float v0 = acc0[i] + bc0;
      float v1 = acc1[i] + bc1;
      float v2 = acc2[i] + bc2;
      float v3 = acc3[i] + bc3;
      if (act == ACT_RELU) {
        v0 = fmaxf(v0, 0.0f); v1 = fmaxf(v1, 0.0f);
        v2 = fmaxf(v2, 0.0f); v3 = fmaxf(v3, 0.0f);
      } else if (act == ACT_SIG) {
        v0 = 1.0f / (1.0f + __expf(-v0)); v1 = 1.0f / (1.0f + __expf(-v1));
        v2 = 1.0f / (1.0f + __expf(-v2)); v3 = 1.0f / (1.0f + __expf(-v3));
      }
      float* o = out + (size_t)m * HDIM;
      o[r]      = v0;
      o[16 + r] = v1;
      o[32 + r] = v2;
      o[48 + r] = v3;
    }
  }
}

// ---------------- coupling: xo = xo * exp(s) + t ; ldj += sum(s) ----------------
__global__ void k_couple(float* __restrict__ xo, const float* __restrict__ s,
                         const float* __restrict__ t, float* __restrict__ ldj, int n) {
  __shared__ float red[256];
  int tid = blockIdx.x * blockDim.x + threadIdx.x;
  float sv = 0.0f;
  if (tid < n * HDIM) {
    sv = s[tid];
    xo[tid] = xo[tid] * __expf(sv) + t[tid];
  }
  red[threadIdx.x] = sv;
  __syncthreads();
  for (int off = 128; off > 0; off >>= 1) {
    if ((int)threadIdx.x < off) red[threadIdx.x] += red[threadIdx.x + off];
    __syncthreads();
  }
  if (threadIdx.x == 0) atomicAdd(ldj, red[0]);
}

// ---------------- final pack: out[N x 128] = [x0 | x1] ----------------
__global__ void k_pack(const float* __restrict__ x0, const float* __restrict__ x1,
                       float* __restrict__ out, int n) {
  int tid = blockIdx.x * blockDim.x + threadIdx.x;
  if (tid >= n * 2 * HDIM) return;
  int node = tid >> 7, c = tid & 127;
  out[tid] = (c < HDIM) ? x0[(size_t)node * HDIM + c]
                        : x1[(size_t)node * HDIM + (c - HDIM)];
}

extern "C" void kernel_launch(void* const* d_in, const int* in_sizes, int n_in,
                              void* d_out, int out_size, void* d_ws, size_t ws_size,
                              hipStream_t stream) {
  (void)in_sizes; (void)n_in; (void)out_size; (void)ws_size;

  const float* x  = (const float*)d_in[0];
  const int*   ei = (const int*)d_in[1];   // (2, E) row-major: [src | dst]
  const float* W  = (const float*)d_in[2]; // (8, 2, 64, 64)
  const float* b  = (const float*)d_in[3]; // (8, 2, 64)
  float* out = (float*)d_out;

  const int n = NN, e = EE;
  const int* src = ei;
  const int* dst = ei + e;

  float* ws   = (float*)d_ws;
  float* dinv = ws;                 ws += n;
  float* x0   = ws;                 ws += (size_t)n * HDIM;
  float* x1   = ws;                 ws += (size_t)n * HDIM;
  float* P    = ws;                 ws += (size_t)n * HDIM;
  float* Hs   = ws;                 ws += (size_t)n * HDIM;
  float* Ht   = ws;                 ws += (size_t)n * HDIM;
  float* Ps   = ws;                 ws += (size_t)n * HDIM;
  float* Pt   = ws;                 ws += (size_t)n * HDIM;
  float* ldj  = out + (size_t)n * 2 * HDIM;  // d_out = [N*128 floats | ldj]

  const dim3 blk(256);
  const int gNF  = (n * HDIM + 255) / 256;
  const int gN   = (n + 255) / 256;
  const int gE   = (e + 255) / 256;
  const int gE16 = (e * 16 + 255) / 256;
  const int gG   = (n + 127) / 128;
  const int gP   = (n * 2 * HDIM + 255) / 256;

  // normalization setup
  k_split_x<<<gNF, blk, 0, stream>>>(x, x0, x1, n);
  k_deg_init<<<gN, blk, 0, stream>>>(dinv, ldj, n);
  k_deg_edges<<<gE, blk, 0, stream>>>(dst, dinv, e);
  k_dinv<<<gN, blk, 0, stream>>>(dinv, n);

  auto prop = [&](const float* h, float* Pout) {
    k_prop_self<<<gNF, blk, 0, stream>>>(h, dinv, Pout, n);
    k_prop_edges<<<gE16, blk, 0, stream>>>(h, src, dst, dinv, Pout, e);
  };
  auto Wl = [&](int g, int l) { return W + (size_t)(g * 2 + l) * HDIM * HDIM; };
  auto Bl = [&](int g, int l) { return b + (size_t)(g * 2 + l) * HDIM; };

  const int T = 2;
  for (int i = 0; i < T; i++) {
    for (int half = 0; half < 2; half++) {
      float* xin  = (half == 0) ? x0 : x1;   // conditioning half
      float* xupd = (half == 0) ? x1 : x0;   // half being transformed
      int gs = half * 2 * T + 0 * T + i;     // s-GCN id
      int gt = half * 2 * T + 1 * T + i;     // t-GCN id

      // Layer 1: A(xW) = (Ax)W -> one shared propagation for both s and t GCNs
      prop(xin, P);
      k_gemm64<<<gG, blk, 0, stream>>>(P, Wl(gs, 0), Bl(gs, 0), Hs, n, ACT_RELU);
      k_gemm64<<<gG, blk, 0, stream>>>(P, Wl(gt, 0), Bl(gt, 0), Ht, n, ACT_RELU);
      // Layer 2
      prop(Hs, Ps);
      prop(Ht, Pt);
      k_gemm64<<<gG, blk, 0, stream>>>(Ps, Wl(gs, 1), Bl(gs, 1), Hs, n, ACT_SIG); // s
      k_gemm64<<<gG, blk, 0, stream>>>(Pt, Wl(gt, 1), Bl(gt, 1), Ht, n, ACT_SIG); // t
      // affine coupling + logdet accumulation
      k_couple<<<gNF, blk, 0, stream>>>(xupd, Hs, Ht, ldj, n);
    }
  }
  k_pack<<<gP, blk, 0, stream>>>(x0, x1, out, n);
}